// AttentionLayer_5952824672393
// MI455X (gfx1250) — compile-verified
//
#include <hip/hip_runtime.h>

// ---------------------------------------------------------------------------
// Reference collapse: softmax over a size-1 axis => alphas == 1, so
//   out[b, 0, d] = sum_{t=0..TX-1} a[b, t, d]
// Implemented as ones(1xTX) @ a[b] via V_WMMA_F32_16X16X4_F32 (f32, lossless),
// streaming a[] exactly once (128 MiB) -> bandwidth-bound (~5.8 us @ 23.3 TB/s).
// ---------------------------------------------------------------------------

typedef __attribute__((ext_vector_type(2))) float v2f;
typedef __attribute__((ext_vector_type(8))) float v8f;

#define B_DIM 128
#define TX    512
#define DA    512

__global__ __launch_bounds__(256) void attn_ctx_colsum_wmma(
    const float* __restrict__ a, float* __restrict__ out)
{
    const int lane  = threadIdx.x & 31;
    const int wib   = threadIdx.x >> 5;                 // wave in block, 0..7
    const int gw    = blockIdx.x * 8 + wib;             // global wave, 0..4095
    const int b     = gw >> 5;                          // batch, 0..127
    const int d0    = (gw & 31) << 4;                   // d-chunk base, 0..496

    const int half  = lane >> 4;                        // lane half: 0 or 1
    const int dn    = lane & 15;                        // N index within tile

    // This lane's column pointer; lane-half selects which pair of t-rows it
    // supplies (.x / .y of the B operand). A = ones makes the reduction
    // invariant to the exact K<->(vgpr,half) mapping: D = C + sum_k B[k, n].
    const float* p = a + ((size_t)b * TX + (size_t)(2 * half)) * DA + d0 + dn;

#if __has_builtin(__builtin_amdgcn_wmma_f32_16x16x4_f32)
    v2f A; A.x = 1.0f; A.y = 1.0f;                      // all-ones A matrix
    v8f c = {};                                         // f32 accumulator tile

    #pragma unroll 4
    for (int t = 0; t < TX; t += 4) {
        v2f Bm;
        Bm.x = p[(size_t)t * DA];                       // row t + 2*half
        Bm.y = p[(size_t)t * DA + DA];                  // row t + 2*half + 1
        // 8 args: (neg_a, A, neg_b, B, c_mod, C, reuse_a, reuse_b)
        c = __builtin_amdgcn_wmma_f32_16x16x4_f32(
                false, A, false, Bm, (short)0, c, false, false);
    }

    // Every D row is identical (A == ones); row M=0 lives in c[0] lanes 0..15.
    if (lane < 16) {
        out[b * DA + d0 + dn] = c[0];
    }
#else
    // Fallback (should not trigger per probe): plain VALU reduction, same math.
    float s0 = 0.0f, s1 = 0.0f;
    #pragma unroll 4
    for (int t = 0; t < TX; t += 4) {
        s0 += p[(size_t)t * DA];
        s1 += p[(size_t)t * DA + DA];
    }
    float s = s0 + s1;
    // Each (half, dn) covered twice; combine via LDS-free trick: shuffle add.
    s += __shfl_xor(s, 16, 32);
    if (lane < 16) out[b * DA + d0 + dn] = s;
#endif
}

extern "C" void kernel_launch(void* const* d_in, const int* in_sizes, int n_in,
                              void* d_out, int out_size, void* d_ws, size_t ws_size,
                              hipStream_t stream)
{
    (void)in_sizes; (void)n_in; (void)d_ws; (void)ws_size; (void)out_size;
    const float* a = (const float*)d_in[0];   // [128, 512, 512] f32
    float* out = (float*)d_out;               // [128, 1, 512]   f32

    // 4096 waves total = 512 blocks x 8 waves; exact cover, no bounds checks.
    attn_ctx_colsum_wmma<<<dim3(512), dim3(256), 0, stream>>>(a, out);
}